// ChowLiuTreePC_40458591928463
// MI455X (gfx1250) — compile-verified
//
#include <hip/hip_runtime.h>
#include <hip/hip_bf16.h>
#include <math.h>

typedef __attribute__((ext_vector_type(2))) float v2f;
typedef __attribute__((ext_vector_type(8))) float v8f;

#define HALF_LOG_2PI 0.91893853320467274178f
#define SPLITK 4

// ---------------- coefficient preprocessing (O(D), negligible) ----------------

__global__ void zero_deg_kernel(int* __restrict__ deg, int D) {
    int i = blockIdx.x * blockDim.x + threadIdx.x;
    if (i < D) deg[i] = 0;
}

__global__ void deg_kernel(const int* __restrict__ edges, int n_end,
                           int* __restrict__ deg, int D) {
    int i = blockIdx.x * blockDim.x + threadIdx.x;
    if (i < n_end) {
        int k = edges[i];
        if ((unsigned)k < (unsigned)D) atomicAdd(&deg[k], 1);  // integer atomic: deterministic
    }
}

__global__ void coeff_kernel(const float* __restrict__ raw, const int* __restrict__ deg,
                             float* __restrict__ g, float* __restrict__ m,
                             float* __restrict__ h, int D) {
    int k = blockIdx.x * blockDim.x + threadIdx.x;
    if (k < D) {
        float mean = 2.0f * tanhf(raw[k]);
        float p = raw[D + k];
        // softplus(p) = max(p,0) + log1p(exp(-|p|)), overflow-safe
        float sd = fmaxf(p, 0.0f) + log1pf(expf(-fabsf(p))) + 1e-6f;
        float dk = (float)deg[k];
        g[k] = -0.5f * dk / (sd * sd);
        m[k] = mean;
        h[k] = -dk * (logf(sd) + HALF_LOG_2PI);
    }
}

// deterministic single-block tree reduction of h -> scalar H
__global__ void hsum_kernel(const float* __restrict__ h, int D, float* __restrict__ Hout) {
    __shared__ float s[256];
    int tid = threadIdx.x;
    float a = 0.0f;
    for (int k = tid; k < D; k += 256) a += h[k];
    s[tid] = a;
    __syncthreads();
    for (int off = 128; off > 0; off >>= 1) {
        if (tid < off) s[tid] += s[tid + off];
        __syncthreads();
    }
    if (tid == 0) Hout[0] = s[0];
}

// ---------------- main streaming pass: WMMA as 16-row x K weighted reducer ----------------
// out_partial[row, seg] = sum_{k in seg} g_k * (x[row,k]-m_k)^2
// One wave owns 16 rows x (D/SPLITK) features. A = g*(x-m)^2 (16x4 f32 tile),
// B = ones(4x16)  =>  every column of D = rowsum(A); accumulate in C across K.
// Loop control is scalarized via readfirstlane (wave-uniform by construction,
// keeps EXEC all-ones around v_wmma), unrolled 4x with two independent C chains.

__device__ __forceinline__ v8f wmma_acc_step(const float* xp, const float* mp,
                                             const float* gp, v2f bones, v8f c) {
    v2f xv = *(const v2f*)xp;          // global_load_b64, 8B-aligned
    v2f mv = *(const v2f*)mp;
    v2f gv = *(const v2f*)gp;
    float t0 = xv.x - mv.x;
    float t1 = xv.y - mv.y;
    v2f A;
    A.x = gv.x * t0 * t0;
    A.y = gv.y * t1 * t1;
    return __builtin_amdgcn_wmma_f32_16x16x4_f32(
        /*neg_a=*/false, A, /*neg_b=*/false, bones,
        /*c_mod=*/(short)0, c, /*reuse_a=*/false, /*reuse_b=*/false);
}

__global__ void __launch_bounds__(256)
chowliu_wmma_kernel(const float* __restrict__ x,
                    const float* __restrict__ g,
                    const float* __restrict__ m,
                    float* __restrict__ partials,
                    int N, int D, int G16, int segLen)
{
    int tid = threadIdx.x;
    // wave-uniform scalars, forced into SGPRs -> scalar loop control
    int gw = __builtin_amdgcn_readfirstlane(blockIdx.x * (blockDim.x >> 5) + (tid >> 5));
    int rowGroup = gw / SPLITK;
    int seg      = gw - rowGroup * SPLITK;
    if (rowGroup >= G16) return;                               // scalar branch

    int lane = tid & 31;
    int r    = lane & 15;           // A-tile row (M) held by this lane
    int hs   = lane >> 4;           // 0: K={0,1}; 1: K={2,3} within the 4-wide chunk

    long long row  = (long long)rowGroup * 16 + r;
    long long rowc = (row < N) ? row : (long long)(N - 1);     // clamp reads; writes guarded
    const float* __restrict__ xrow = x + rowc * (long long)D;

    int ks = seg * segLen;
    int ke = ks + segLen;
    if (ke > D) ke = D;

    v8f c0 = {0.f, 0.f, 0.f, 0.f, 0.f, 0.f, 0.f, 0.f};
    v8f c1 = {0.f, 0.f, 0.f, 0.f, 0.f, 0.f, 0.f, 0.f};
    v2f bones; bones.x = 1.0f; bones.y = 1.0f;                 // B = ones (layout-invariant)

    if (ks < ke) __builtin_prefetch(xrow + ks, 0, 1);          // global_prefetch_b8

    int off2 = 2 * hs;                                         // lane's K-pair within chunk
    int k0 = ks;
    // 4x unrolled main body: 16 K per trip, immediate offsets off shared bases
    for (; k0 + 15 < ke; k0 += 16) {
        const float* xp = xrow + k0 + off2;
        const float* mp = m    + k0 + off2;
        const float* gp = g    + k0 + off2;
        c0 = wmma_acc_step(xp,      mp,      gp,      bones, c0);
        c1 = wmma_acc_step(xp + 4,  mp + 4,  gp + 4,  bones, c1);
        c0 = wmma_acc_step(xp + 8,  mp + 8,  gp + 8,  bones, c0);
        c1 = wmma_acc_step(xp + 12, mp + 12, gp + 12, bones, c1);
    }
    // remainder full 4-K chunks
    for (; k0 + 3 < ke; k0 += 4) {
        c0 = wmma_acc_step(xrow + k0 + off2, m + k0 + off2, g + k0 + off2, bones, c0);
    }
    if (k0 < ke) {                                             // uniform tail, 1..3 features
        int k = k0 + off2;
        float x0 = (k     < ke) ? xrow[k]     : 0.0f;
        float x1 = (k + 1 < ke) ? xrow[k + 1] : 0.0f;
        float m0 = (k     < ke) ? m[k]        : 0.0f;
        float m1 = (k + 1 < ke) ? m[k + 1]    : 0.0f;
        float g0 = (k     < ke) ? g[k]        : 0.0f;
        float g1 = (k + 1 < ke) ? g[k + 1]    : 0.0f;
        float t0 = x0 - m0, t1 = x1 - m1;
        v2f A;
        A.x = g0 * t0 * t0;
        A.y = g1 * t1 * t1;
        c0 = __builtin_amdgcn_wmma_f32_16x16x4_f32(
                false, A, false, bones, (short)0, c0, false, false);
    }

    v8f c = c0 + c1;                                           // merge accumulator chains

    // C/D layout: lane L holds column N=L&15 of rows M = vgpr + 8*(L>>4).
    // All columns are identical (B=ones). Lanes with (L&15)==v on each half
    // write row v + 8*hs from accumulator element v (v in 0..7).
    int v = r;
    if (v < 8) {
        float val = c[0];
        if (v == 1) val = c[1];
        if (v == 2) val = c[2];
        if (v == 3) val = c[3];
        if (v == 4) val = c[4];
        if (v == 5) val = c[5];
        if (v == 6) val = c[6];
        if (v == 7) val = c[7];
        long long orow = (long long)rowGroup * 16 + v + 8 * hs;
        if (orow < N) partials[orow * SPLITK + seg] = val;
    }
}

__global__ void finish_kernel(const float* __restrict__ partials,
                              const float* __restrict__ Hp,
                              float* __restrict__ out, int N) {
    int i = blockIdx.x * blockDim.x + threadIdx.x;
    if (i < N) {
        const float* p = partials + (size_t)i * SPLITK;
        out[i] = p[0] + p[1] + p[2] + p[3] + Hp[0];
    }
}

// ---------------- launch ----------------

extern "C" void kernel_launch(void* const* d_in, const int* in_sizes, int n_in,
                              void* d_out, int out_size, void* d_ws, size_t ws_size,
                              hipStream_t stream) {
    const float* x     = (const float*)d_in[0];
    const float* raw   = (const float*)d_in[1];
    const int*   edges = (const int*)d_in[2];
    float*       out   = (float*)d_out;

    int D = in_sizes[1] / 2;                       // raw_params = [2D]
    long long ND = (long long)in_sizes[0];
    int N = (int)(ND / D);
    int nEnd = in_sizes[2];                        // 2*E endpoint entries

    // workspace layout
    float* g        = (float*)d_ws;
    float* m        = g + D;
    float* h        = m + D;
    int*   deg      = (int*)(h + D);
    float* H        = (float*)(deg + D);
    float* partials = H + 16;                      // keep 16B alignment cushion

    zero_deg_kernel<<<(D + 255) / 256, 256, 0, stream>>>(deg, D);
    deg_kernel<<<(nEnd + 255) / 256, 256, 0, stream>>>(edges, nEnd, deg, D);
    coeff_kernel<<<(D + 255) / 256, 256, 0, stream>>>(raw, deg, g, m, h, D);
    hsum_kernel<<<1, 256, 0, stream>>>(h, D, H);

    int G16 = (N + 15) / 16;
    int segLen = (((D + SPLITK - 1) / SPLITK) + 3) & ~3;       // 4-aligned K segment
    long long totalWaves = (long long)G16 * SPLITK;
    int blocks = (int)((totalWaves + 7) / 8);                  // 8 waves / 256-thread block
    chowliu_wmma_kernel<<<blocks, 256, 0, stream>>>(x, g, m, partials, N, D, G16, segLen);

    finish_kernel<<<(N + 255) / 256, 256, 0, stream>>>(partials, H, out, N);
}